// CBOWSubword_88330297409633
// MI455X (gfx1250) — compile-verified
//
#include <hip/hip_runtime.h>
#include <stdint.h>
#include <stddef.h>

// CBOW-subword embedding: out[t] = W[seq[t]] + W[pre[t]] + W[post[t]]
// B*S = 131072 tokens, D = 128 floats (512 B per row).
// Memory-bound: ~117 MB of HBM traffic after the 51 MB table settles in L2.

#define EMBED 128   // floats per embedding row
#define UNROLL 4    // tokens per wave

typedef float f4 __attribute__((ext_vector_type(4)));

// ---------------------------------------------------------------------------
// Primary kernel (launched): direct-to-VGPR gather-sum.
//   - wave-uniform token base -> scalar (SMEM) index loads
//   - per-lane float4 row loads (coalesced 512 B per wave per row), default TH
//     so weight rows stay resident in the 192 MB L2
//   - non-temporal b128 stores so the 64 MB output stream does not evict the
//     weight table from L2
// ---------------------------------------------------------------------------
__global__ __launch_bounds__(256)
void cbow3_gather_sum(const int* __restrict__ seq,
                      const int* __restrict__ pre,
                      const int* __restrict__ post,
                      const float* __restrict__ W,
                      float* __restrict__ out,
                      int n_tokens)
{
    const int lane = threadIdx.x & 31;
    int wv = blockIdx.x * (blockDim.x >> 5) + (threadIdx.x >> 5);
    wv = __builtin_amdgcn_readfirstlane(wv);      // wave-uniform -> SGPR
    const int t0 = wv * UNROLL;
    if (t0 >= n_tokens) return;

    const int col = lane << 2;                    // float offset inside row

    if (t0 + UNROLL <= n_tokens) {
        // ---- full tile: 12 scalar index loads, 12 vector b128 gathers ----
        int ia[UNROLL], ib[UNROLL], ic[UNROLL];
#pragma unroll
        for (int u = 0; u < UNROLL; ++u) {
            ia[u] = seq [t0 + u];
            ib[u] = pre [t0 + u];
            ic[u] = post[t0 + u];
        }
        f4 a[UNROLL], b[UNROLL], c[UNROLL];
#pragma unroll
        for (int u = 0; u < UNROLL; ++u)
            a[u] = *(const f4*)(W + (size_t)ia[u] * EMBED + col);
#pragma unroll
        for (int u = 0; u < UNROLL; ++u)
            b[u] = *(const f4*)(W + (size_t)ib[u] * EMBED + col);
#pragma unroll
        for (int u = 0; u < UNROLL; ++u)
            c[u] = *(const f4*)(W + (size_t)ic[u] * EMBED + col);
#pragma unroll
        for (int u = 0; u < UNROLL; ++u) {
            f4 r = a[u] + b[u] + c[u];
            __builtin_nontemporal_store(
                r, (f4*)(out + (size_t)(t0 + u) * EMBED + col));
        }
    } else {
        // ---- ragged tail (not hit for 131072 tokens, kept for generality) ----
        for (int t = t0; t < n_tokens; ++t) {
            const int i0 = seq[t], i1 = pre[t], i2 = post[t];
            f4 a = *(const f4*)(W + (size_t)i0 * EMBED + col);
            f4 b = *(const f4*)(W + (size_t)i1 * EMBED + col);
            f4 c = *(const f4*)(W + (size_t)i2 * EMBED + col);
            f4 r = a + b + c;
            __builtin_nontemporal_store(
                r, (f4*)(out + (size_t)t * EMBED + col));
        }
    }
}

// ---------------------------------------------------------------------------
// Probe kernel (compiled, NOT launched): same gather routed through the
// CDNA5 async Global->LDS path. Purpose this round: verify that
// global_load_async_to_lds_b128 / s_wait_asynccnt assemble for gfx1250 and
// appear in the histogram; a later round can adopt this as the launched path
// if the async pipelining proves worth the LDS round-trip.
// ---------------------------------------------------------------------------
__global__ __launch_bounds__(256)
void cbow3_gather_sum_async(const int* __restrict__ seq,
                            const int* __restrict__ pre,
                            const int* __restrict__ post,
                            const float* __restrict__ W,
                            float* __restrict__ out,
                            int n_tokens)
{
    __shared__ float tile[8][3][UNROLL][EMBED];   // 48 KB / workgroup

    const int lane = threadIdx.x & 31;
    const int wib  = (threadIdx.x >> 5) & 7;      // wave-in-block
    int wv = blockIdx.x * (blockDim.x >> 5) + (threadIdx.x >> 5);
    wv = __builtin_amdgcn_readfirstlane(wv);
    const int t0 = wv * UNROLL;
    if (t0 + UNROLL > n_tokens) return;           // probe: full tiles only

    const int col = lane << 2;

    int ia[UNROLL], ib[UNROLL], ic[UNROLL];
#pragma unroll
    for (int u = 0; u < UNROLL; ++u) {
        ia[u] = seq [t0 + u];
        ib[u] = pre [t0 + u];
        ic[u] = post[t0 + u];
    }

#pragma unroll
    for (int u = 0; u < UNROLL; ++u) {
        const float* ga = W + (size_t)ia[u] * EMBED + col;
        const float* gb = W + (size_t)ib[u] * EMBED + col;
        const float* gc = W + (size_t)ic[u] * EMBED + col;
        // generic shared-pointer low 32 bits == LDS byte address (flat aperture)
        uint32_t la = (uint32_t)(uintptr_t)&tile[wib][0][u][col];
        uint32_t lb = (uint32_t)(uintptr_t)&tile[wib][1][u][col];
        uint32_t lc = (uint32_t)(uintptr_t)&tile[wib][2][u][col];
        asm volatile("global_load_async_to_lds_b128 %0, %1, off"
                     :: "v"(la), "v"(ga) : "memory");
        asm volatile("global_load_async_to_lds_b128 %0, %1, off"
                     :: "v"(lb), "v"(gb) : "memory");
        asm volatile("global_load_async_to_lds_b128 %0, %1, off"
                     :: "v"(lc), "v"(gc) : "memory");
    }
    asm volatile("s_wait_asynccnt 0x0" ::: "memory");

#pragma unroll
    for (int u = 0; u < UNROLL; ++u) {
        f4 a = *(const f4*)&tile[wib][0][u][col];
        f4 b = *(const f4*)&tile[wib][1][u][col];
        f4 c = *(const f4*)&tile[wib][2][u][col];
        f4 r = a + b + c;
        __builtin_nontemporal_store(
            r, (f4*)(out + (size_t)(t0 + u) * EMBED + col));
    }
}

// ---------------------------------------------------------------------------
extern "C" void kernel_launch(void* const* d_in, const int* in_sizes, int n_in,
                              void* d_out, int out_size, void* d_ws, size_t ws_size,
                              hipStream_t stream)
{
    const int*   seq  = (const int*)  d_in[0];   // [B,S] int32
    const int*   pre  = (const int*)  d_in[1];   // [B,S] int32
    const int*   post = (const int*)  d_in[2];   // [B,S] int32
    const float* W    = (const float*)d_in[3];   // [VOCAB,128] f32
    float*       out  = (float*)      d_out;     // [B,S,128] f32

    const int n_tokens = in_sizes[0];            // B*S = 131072

    const int waves   = (n_tokens + UNROLL - 1) / UNROLL;
    const int threads = waves * 32;
    const int block   = 256;                     // 8 waves / workgroup
    const int grid    = (threads + block - 1) / block;

    cbow3_gather_sum<<<grid, block, 0, stream>>>(seq, pre, post, W, out, n_tokens);
}